// ThresholdRateEncoder_28922309771718
// MI455X (gfx1250) — compile-verified
//
#include <hip/hip_runtime.h>
#include <stdint.h>

// LIF spike encoder: v = 0.9*v + u_t; s = (v>0); v -= s, per row over T steps.
// Memory-bound (512 MB traffic, ~22us floor at 23.3 TB/s). Strategy:
//  - wave-cooperative coalesced async copies global->LDS (b128, 4 rows/instr)
//  - per-lane sequential recurrence out of LDS (low latency, padded stride)
//  - spikes staged in LDS, async-stored LDS->global coalesced
//  - double-buffered, pipelined with s_wait_asynccnt thresholds (wave32)
//  - non-temporal hints: both streams are touched exactly once
//  - __fmul_rn/__fadd_rn to match the reference's unfused mul+add exactly
//    (an FMA's 1-ulp difference can flip a thresholded spike and corrupt
//     the rest of that row's sequential scan)

#define T_LEN        1024
#define CT           32                 // timesteps per chunk (128 B per row)
#define NC           (T_LEN / CT)       // 32 chunks
#define ROW_STRIDE   36                 // dwords; 144 B = 16B-aligned, low bank conflict
#define WAVES_PER_BLOCK 8
#define ROWS_PER_WAVE   32
#define ROWS_PER_BLOCK  (WAVES_PER_BLOCK * ROWS_PER_WAVE)

#define LEAK 0.9f

__global__ __launch_bounds__(256) void lif_async_kernel(
    const float* __restrict__ u, float* __restrict__ out) {

  __shared__ float sIn [WAVES_PER_BLOCK][2][ROWS_PER_WAVE * ROW_STRIDE];
  __shared__ float sOut[WAVES_PER_BLOCK][2][ROWS_PER_WAVE * ROW_STRIDE];

  const int lane = threadIdx.x & 31;
  const int wave = threadIdx.x >> 5;
  const int waveRow0 = blockIdx.x * ROWS_PER_BLOCK + wave * ROWS_PER_WAVE;

  // Cooperative transfer mapping: each b128 instruction moves 4 rows x 128B.
  // lane -> (row-in-group-of-4, 16B sub-chunk 0..7)
  const int rSub = lane >> 3;   // 0..3
  const int cSub = lane & 7;    // 0..7

  // LDS byte offsets (low 32 bits of generic pointer == LDS offset on CDNA5)
  const uint32_t inBase [2] = { (uint32_t)(uintptr_t)&sIn [wave][0][0],
                                (uint32_t)(uintptr_t)&sIn [wave][1][0] };
  const uint32_t outBase[2] = { (uint32_t)(uintptr_t)&sOut[wave][0][0],
                                (uint32_t)(uintptr_t)&sOut[wave][1][0] };

  auto issue_loads = [&](int c, uint32_t ldsBase) {
#pragma unroll
    for (int i = 0; i < 8; ++i) {
      const int rLocal = i * 4 + rSub;
      const int r = waveRow0 + rLocal;
      const float* g = u + (size_t)r * T_LEN + c * CT + cSub * 4;
      const uint32_t l = ldsBase + (uint32_t)(rLocal * ROW_STRIDE * 4 + cSub * 16);
      asm volatile("global_load_async_to_lds_b128 %0, %1, off th:TH_LOAD_NT"
                   :: "v"(l), "v"(g) : "memory");
    }
  };

  auto issue_stores = [&](int c, uint32_t ldsBase) {
#pragma unroll
    for (int i = 0; i < 8; ++i) {
      const int rLocal = i * 4 + rSub;
      const int r = waveRow0 + rLocal;
      float* g = out + (size_t)r * T_LEN + c * CT + cSub * 4;
      const uint32_t l = ldsBase + (uint32_t)(rLocal * ROW_STRIDE * 4 + cSub * 16);
      asm volatile("global_store_async_from_lds_b128 %0, %1, off th:TH_STORE_NT"
                   :: "v"(g), "v"(l) : "memory");
    }
  };

  // Prologue: prefetch two chunks (16 outstanding async loads).
  issue_loads(0, inBase[0]);
  issue_loads(1, inBase[1]);

  float v = 0.0f;

  for (int c = 0; c < NC; ++c) {
    // Loads complete in-order among loads; stores in-order among stores.
    // With <=16 loads + <=8 stores outstanding, cnt<=8 guarantees:
    //   (1) all 8 loads of chunk c have landed (else 9+ would be pending)
    //   (2) all stores of chunk c-2 drained (else 9+ would be pending)
    asm volatile("s_wait_asynccnt 0x8" ::: "memory");

    const int b = c & 1;
    const float* __restrict__ in = &sIn [wave][b][0];
    float* __restrict__ o        = &sOut[wave][b][0];

#pragma unroll
    for (int t = 0; t < CT; ++t) {
      const float x = in[lane * ROW_STRIDE + t];
      v = __fadd_rn(__fmul_rn(LEAK, v), x);     // no FMA contraction
      const float s = (v > 0.0f) ? 1.0f : 0.0f;
      v = __fadd_rn(v, -s);
      o[lane * ROW_STRIDE + t] = s;
    }

    // Make LDS spike writes visible to the async store engine.
    asm volatile("s_wait_dscnt 0x0" ::: "memory");
    issue_stores(c, outBase[b]);

    if (c + 2 < NC) issue_loads(c + 2, inBase[b]);
  }
  // s_endpgm performs an implicit wait-idle; all async stores drain.
}

// Fallback for any row remainder (not expected for the reference shape):
// one thread per row, simple sequential scan.
__global__ void lif_tail_kernel(const float* __restrict__ u,
                                float* __restrict__ out,
                                int rowStart, int rows) {
  const int r = rowStart + blockIdx.x * blockDim.x + threadIdx.x;
  if (r >= rows) return;
  const float* __restrict__ ur = u + (size_t)r * T_LEN;
  float* __restrict__ orow = out + (size_t)r * T_LEN;
  float v = 0.0f;
  for (int t = 0; t < T_LEN; ++t) {
    v = __fadd_rn(__fmul_rn(LEAK, v), ur[t]);
    const float s = (v > 0.0f) ? 1.0f : 0.0f;
    v = __fadd_rn(v, -s);
    orow[t] = s;
  }
}

extern "C" void kernel_launch(void* const* d_in, const int* in_sizes, int n_in,
                              void* d_out, int out_size, void* d_ws, size_t ws_size,
                              hipStream_t stream) {
  const float* u = (const float*)d_in[0];
  float* out = (float*)d_out;

  const int total = in_sizes[0];          // b*C*T = 16*4096*1024
  const int rows  = total / T_LEN;        // 65536
  const int fullBlocks = rows / ROWS_PER_BLOCK;

  if (fullBlocks > 0) {
    lif_async_kernel<<<fullBlocks, 256, 0, stream>>>(u, out);
  }
  const int done = fullBlocks * ROWS_PER_BLOCK;
  const int rem = rows - done;
  if (rem > 0) {
    const int tb = 256;
    lif_tail_kernel<<<(rem + tb - 1) / tb, tb, 0, stream>>>(u, out, done, rows);
  }
}